// TopQPooling_51745765982324
// MI455X (gfx1250) — compile-verified
//
#include <hip/hip_runtime.h>
#include <hip/hip_bf16.h>

#define BB 16
#define TT 4096
#define DD 1024
#define KMAX 640   // >= ceil(0.15*4096)=615, padded
#define SPLIT 4    // K-dimension split for the WMMA gather pass

typedef __attribute__((ext_vector_type(2))) float v2f;
typedef __attribute__((ext_vector_type(8))) float v8f;

// ---------------------------------------------------------------------------
// Pass 1: streaming sum-of-squares per row -> monotone uint key (0 if masked)
// One wave per row; lane loads 8 float4s (coalesced 512B per wave per step).
// 256 MB @ 23.3 TB/s ~= 11 us: this pass is the roofline floor.
// ---------------------------------------------------------------------------
__global__ void topq_norm_keys(const float* __restrict__ H,
                               const unsigned char* __restrict__ mask,
                               unsigned int* __restrict__ keys) {
    const int wid  = threadIdx.x >> 5;
    const int lane = threadIdx.x & 31;
    const int row  = blockIdx.x * 8 + wid;          // 0 .. B*T-1
    const float4* p = (const float4*)(H + (size_t)row * DD);
    // gfx1250 prefetch of the next row this wave-slot will touch
    __builtin_prefetch((const void*)(p + 8 * 256), 0, 0);
    float ssq = 0.f;
#pragma unroll
    for (int i = 0; i < 8; ++i) {
        float4 v = p[i * 32 + lane];
        ssq += v.x * v.x + v.y * v.y + v.z * v.z + v.w * v.w;
    }
    for (int off = 16; off; off >>= 1) ssq += __shfl_xor(ssq, off, 32);
    if (lane == 0) {
        keys[row] = mask[row] ? __float_as_uint(ssq) : 0u;
    }
}

// ---------------------------------------------------------------------------
// Pass 2: K = max_b max(1, ceil(0.15 * valid_len_b)). One wave per batch.
// ---------------------------------------------------------------------------
__global__ void topq_kval(const unsigned char* __restrict__ mask,
                          int* __restrict__ Kval) {
    __shared__ int kb[BB];
    const int wid  = threadIdx.x >> 5;
    const int lane = threadIdx.x & 31;
    const unsigned int* m = (const unsigned int*)(mask + wid * TT);
    int s = 0;
    for (int i = lane; i < TT / 4; i += 32) {
        unsigned int x = m[i];
        s += (int)((x & 0xFFu) + ((x >> 8) & 0xFFu) + ((x >> 16) & 0xFFu) + (x >> 24));
    }
    for (int off = 16; off; off >>= 1) s += __shfl_xor(s, off, 32);
    if (lane == 0) kb[wid] = max(1, (int)ceilf(0.15f * (float)s));
    __syncthreads();
    if (threadIdx.x == 0) {
        int K = 1;
        for (int i = 0; i < BB; ++i) K = max(K, kb[i]);
        *Kval = K;
    }
}

// ---------------------------------------------------------------------------
// Pass 3: per batch, exact K-th-largest via 32-step MSB radix select over the
// bit-keys in LDS, tie fix-up in index order (matches lax.top_k), then a
// block-wide scan compacts the selected rows (ascending) into idxList.
// Entries are pre-scaled ELEMENT OFFSETS (t * DD) so pass 4 can use cheap
// 32-bit scale_offset addressing; list is zero-padded to KMAX.
// Block = 1024 threads (32 waves); grid = B.
// ---------------------------------------------------------------------------
__global__ void topq_select(const unsigned int* __restrict__ keys,
                            const int* __restrict__ Kval,
                            int* __restrict__ idxList) {
    __shared__ unsigned int skey[TT];   // 16 KB
    __shared__ unsigned int ssel[TT];   // 16 KB
    __shared__ unsigned int red[32];
    __shared__ unsigned int wsum[32];
    __shared__ unsigned int bc;

    const int b    = blockIdx.x;
    const int tid  = threadIdx.x;
    const int lane = tid & 31;
    const int wid  = tid >> 5;

    // zero-pad the offset list (barrier-ordered before the final scatter)
    for (int i = tid; i < KMAX; i += 1024) idxList[b * KMAX + i] = 0;

    for (int i = tid; i < TT; i += 1024) skey[i] = keys[b * TT + i];
    __syncthreads();

    const unsigned int K = (unsigned int)(*Kval);
    unsigned int myk[4];
#pragma unroll
    for (int j = 0; j < 4; ++j) myk[j] = skey[tid * 4 + j];

    unsigned int prefix = 0, want = K;
    for (int bit = 31; bit >= 0; --bit) {
        const unsigned int hi = (prefix >> bit) | 1u;
        unsigned int c = 0;
#pragma unroll
        for (int j = 0; j < 4; ++j) c += ((myk[j] >> bit) == hi) ? 1u : 0u;
        for (int off = 16; off; off >>= 1) c += __shfl_xor(c, off, 32);
        if (lane == 0) red[wid] = c;
        __syncthreads();
        if (tid == 0) {
            unsigned int tot = 0;
            for (int i = 0; i < 32; ++i) tot += red[i];
            bc = tot;
        }
        __syncthreads();
        const unsigned int c1 = bc;        // identical in all threads
        if (c1 >= want) prefix |= (1u << bit);
        else            want   -= c1;
        __syncthreads();
    }
    const unsigned int kth = prefix;       // want = #equal keys to take

    // strictly-greater selected in parallel; equal keys taken lowest-index-first
#pragma unroll
    for (int j = 0; j < 4; ++j) ssel[tid * 4 + j] = (myk[j] > kth) ? 1u : 0u;
    __syncthreads();
    if (tid == 0) {
        unsigned int need = want;
        for (int t = 0; t < TT && need; ++t)
            if (skey[t] == kth) { ssel[t] = 1u; --need; }
    }
    __syncthreads();

    // block-wide exclusive scan over ssel -> compact offsets (ascending index)
    unsigned int loc[4], s = 0;
#pragma unroll
    for (int j = 0; j < 4; ++j) { loc[j] = s; s += ssel[tid * 4 + j]; }
    unsigned int v = s;
    for (int off = 1; off < 32; off <<= 1) {
        unsigned int n = __shfl_up(v, off, 32);
        if (lane >= off) v += n;
    }
    if (lane == 31) wsum[wid] = v;
    __syncthreads();
    if (wid == 0) {
        unsigned int w = wsum[lane];
        unsigned int vv = w;
        for (int off = 1; off < 32; off <<= 1) {
            unsigned int n = __shfl_up(vv, off, 32);
            if (lane >= off) vv += n;
        }
        wsum[lane] = vv - w;               // exclusive wave base
    }
    __syncthreads();
    const unsigned int base = wsum[wid] + (v - s);
#pragma unroll
    for (int j = 0; j < 4; ++j)
        if (ssel[tid * 4 + j])
            idxList[b * KMAX + base + loc[j]] = (tid * 4 + j) * DD;  // pre-scaled
}

// ---------------------------------------------------------------------------
// Pass 4: gather-mean as WMMA partials with split-K.
// V_WMMA_F32_16X16X4_F32, A = 1/K broadcast (all 16 M-rows identical, so the
// K-row->VGPR permutation is irrelevant), B = 4 gathered rows x 16 cols.
// idxList holds pre-scaled element offsets -> gather address is one
// v_add_nc_u32 + scale_offset load (no 64-bit sext/shl/add chain).
// Main loop is branchless; tail masks loaded VALUES with cndmask (list is
// zero-padded so addresses are valid; masked rows contribute exactly 0).
// One wave per (batch, 16-col tile, K-split); EXEC stays all-ones.
// ---------------------------------------------------------------------------
__global__ void topq_wmma_partial(const float* __restrict__ H,
                                  const int* __restrict__ Kval,
                                  const int* __restrict__ idxList,
                                  float* __restrict__ part) {
    const int tile = blockIdx.x;           // 0..63 (16-col tiles of D)
    const int b    = blockIdx.y;           // 0..15
    const int sp   = blockIdx.z;           // 0..SPLIT-1
    const int lane = threadIdx.x;          // 0..31
    const int K    = *Kval;
    const float inv = 1.0f / (float)K;

    const int chunk = (((K + SPLIT - 1) / SPLIT) + 3) & ~3;
    const int start = sp * chunk;
    const int end   = (start + chunk < K) ? (start + chunk) : K;
    const int len   = (end > start) ? (end - start) : 0;
    const int kmain = start + (len & ~3);

    v2f a; a.x = inv; a.y = inv;
    v8f c = {0.f, 0.f, 0.f, 0.f, 0.f, 0.f, 0.f, 0.f};

    const unsigned int col = (unsigned int)(tile * 16 + (lane & 15));
    const int khalf = lane >> 4;           // 0 or 1
    const unsigned int* idx = (const unsigned int*)(idxList + b * KMAX);
    const float* Hb = H + (size_t)b * TT * DD;

    int kb = start;
#pragma unroll 4
    for (; kb < kmain; kb += 4) {          // branchless hot loop
        const unsigned int oA = idx[kb + khalf] + col;      // u32 elem offset
        const unsigned int oB = idx[kb + 2 + khalf] + col;
        v2f bm;
        bm.x = Hb[oA];
        bm.y = Hb[oB];
        c = __builtin_amdgcn_wmma_f32_16x16x4_f32(
                false, a, false, bm, (short)0, c, false, false);
    }
    if (kb < end) {                        // scalar branch: EXEC unchanged
        const int kA = kb + khalf;
        const int kB = kb + 2 + khalf;
        const float xA = Hb[idx[kA] + col];  // zero-padded -> always in-bounds
        const float xB = Hb[idx[kB] + col];
        v2f bm;
        bm.x = (kA < end) ? xA : 0.f;      // value-mask, not EXEC-mask
        bm.y = (kB < end) ? xB : 0.f;
        c = __builtin_amdgcn_wmma_f32_16x16x4_f32(
                false, a, false, bm, (short)0, c, false, false);
    }
    if (lane < 16) part[((size_t)sp * BB + b) * DD + col] = c[0];
}

// ---------------------------------------------------------------------------
// Pass 5: deterministic split-K combine (fixed summation order, no atomics).
// ---------------------------------------------------------------------------
__global__ void topq_combine(const float* __restrict__ part,
                             float* __restrict__ out) {
    const int i = blockIdx.x * 256 + threadIdx.x;  // 0 .. B*D-1
    float v = 0.f;
#pragma unroll
    for (int s = 0; s < SPLIT; ++s) v += part[(size_t)s * BB * DD + i];
    out[i] = v;
}

// ---------------------------------------------------------------------------
extern "C" void kernel_launch(void* const* d_in, const int* in_sizes, int n_in,
                              void* d_out, int out_size, void* d_ws, size_t ws_size,
                              hipStream_t stream) {
    const float*         H    = (const float*)d_in[0];
    const unsigned char* mask = (const unsigned char*)d_in[1]; // jnp bool = 1 byte
    float*               out  = (float*)d_out;

    char* ws = (char*)d_ws;
    unsigned int* keys    = (unsigned int*)ws;                       // B*T*4   = 256 KB
    int*          Kval    = (int*)(ws + (size_t)BB * TT * 4);        // 4 B (padded)
    int*          idxList = (int*)(ws + (size_t)BB * TT * 4 + 64);   // B*KMAX*4 ~ 40 KB
    float*        part    = (float*)ws;  // SPLIT*B*D*4 = 256 KB, reuses dead keys region

    topq_norm_keys<<<dim3(BB * TT / 8), dim3(256), 0, stream>>>(H, mask, keys);
    topq_kval<<<dim3(1), dim3(BB * 32), 0, stream>>>(mask, Kval);
    topq_select<<<dim3(BB), dim3(1024), 0, stream>>>(keys, Kval, idxList);
    topq_wmma_partial<<<dim3(DD / 16, BB, SPLIT), dim3(32), 0, stream>>>(H, Kval, idxList, part);
    topq_combine<<<dim3(BB * DD / 256), dim3(256), 0, stream>>>(part, out);
}